// GCN_54133767799052
// MI455X (gfx1250) — compile-verified
//
#include <hip/hip_runtime.h>
#include <hip/hip_bf16.h>
#include <math.h>

// ---------------- sizes ----------------
#define HH 64
#define N_COL 100000
#define N_ROW 80000
#define NE 2000000
#define WEIGHT_DECAY 5e-4f
#define WPB 8   // waves per block (wave32)

typedef float v2f __attribute__((ext_vector_type(2)));
typedef float v8f __attribute__((ext_vector_type(8)));

static __device__ __forceinline__ v8f wmma4(v2f a, v2f b, v8f c) {
  // D = A(16x4 f32) x B(4x16 f32) + C(16x16 f32)
  return __builtin_amdgcn_wmma_f32_16x16x4_f32(
      /*neg_a=*/false, a, /*neg_b=*/false, b,
      /*c_mod=*/(short)0, c, /*reuse_a=*/false, /*reuse_b=*/false);
}

// Y[N x 64] = relu(X[N x K] @ W[K x 64] + bias); optionally duplicate into Ydup.
// One wave handles a 16-row tile; 4 WMMA accumulators cover the 64 output cols.
// K is compile-time: main loop is branch-free; tail (K%4 != 0) uses clamped
// loads + cndmask on the A fragment only (0 * finite == 0).
template <int K>
__global__ void gemm_relu64_t(const float* __restrict__ X, const float* __restrict__ W,
                              const float* __restrict__ bias, float* __restrict__ Y,
                              float* __restrict__ Ydup, int Nrows) {
  constexpr int KFULL = (K / 4) * 4;
  int wave = threadIdx.x >> 5;
  int lane = threadIdx.x & 31;
  int tile = blockIdx.x * WPB + wave;
  if (tile * 16 >= Nrows) return;          // wave-uniform; EXEC stays all-1s
  int rb   = tile * 16;
  int half = lane >> 4;                    // 0: K+0/1, 1: K+2/3
  int l16  = lane & 15;
  const float* __restrict__ xrow = X + (size_t)(rb + l16) * K;

  v8f c0 = {}, c1 = {}, c2 = {}, c3 = {};
#pragma unroll 4
  for (int k0 = 0; k0 < KFULL; k0 += 4) {
    int kk = k0 + 2 * half;
    v2f a;
    if constexpr ((K & 7) == 0) {
      float2 av = *(const float2*)(xrow + kk);   // 8B-aligned: kk even, K%8==0
      a.x = av.x; a.y = av.y;
    } else {
      a.x = xrow[kk]; a.y = xrow[kk + 1];
    }
    const float* w0 = W + (size_t)kk * 64 + l16;
    const float* w1 = w0 + 64;
    v2f b0, b1, b2, b3;
    b0.x = w0[0]; b1.x = w0[16]; b2.x = w0[32]; b3.x = w0[48];
    b0.y = w1[0]; b1.y = w1[16]; b2.y = w1[32]; b3.y = w1[48];
    c0 = wmma4(a, b0, c0);
    c1 = wmma4(a, b1, c1);
    c2 = wmma4(a, b2, c2);
    c3 = wmma4(a, b3, c3);
  }
  if constexpr (KFULL < K) {
    int kk = KFULL + 2 * half;
    int ka = (kk     < K) ? kk     : (K - 1);   // clamped, always in-bounds
    int kb = (kk + 1 < K) ? kk + 1 : (K - 1);
    float ta = xrow[ka], tb = xrow[kb];
    v2f a;
    a.x = (kk     < K) ? ta : 0.f;              // mask A only: 0*finite == 0
    a.y = (kk + 1 < K) ? tb : 0.f;
    const float* w0 = W + (size_t)ka * 64 + l16;
    const float* w1 = W + (size_t)kb * 64 + l16;
    v2f b0, b1, b2, b3;
    b0.x = w0[0]; b1.x = w0[16]; b2.x = w0[32]; b3.x = w0[48];
    b0.y = w1[0]; b1.y = w1[16]; b2.y = w1[32]; b3.y = w1[48];
    c0 = wmma4(a, b0, c0);
    c1 = wmma4(a, b1, c1);
    c2 = wmma4(a, b2, c2);
    c3 = wmma4(a, b3, c3);
  }
#pragma unroll
  for (int r = 0; r < 8; ++r) {
    size_t orow = (size_t)(rb + r + 8 * half);
    float v0 = fmaxf(c0[r] + bias[l16],      0.f);
    float v1 = fmaxf(c1[r] + bias[16 + l16], 0.f);
    float v2 = fmaxf(c2[r] + bias[32 + l16], 0.f);
    float v3 = fmaxf(c3[r] + bias[48 + l16], 0.f);
    float* y = Y + orow * 64;
    y[l16] = v0; y[16 + l16] = v1; y[32 + l16] = v2; y[48 + l16] = v3;
    if (Ydup) {
      float* yd = Ydup + orow * 64;
      yd[l16] = v0; yd[16 + l16] = v1; yd[32 + l16] = v2; yd[48 + l16] = v3;
    }
  }
}

// out[i] = dot(X[i, 0:64], a[0:64]) using float4 (b128) loads
__global__ void rowdot64(const float* __restrict__ X, const float* __restrict__ a,
                         float* __restrict__ out, int N) {
  int i = blockIdx.x * blockDim.x + threadIdx.x;
  if (i >= N) return;
  const float4* xr = (const float4*)(X + (size_t)i * 64);
  const float4* a4 = (const float4*)a;
  float s = 0.f;
#pragma unroll
  for (int k = 0; k < 16; ++k) {
    float4 x = xr[k], w = a4[k];
    s += x.x * w.x + x.y * w.y + x.z * w.z + x.w * w.w;
  }
  out[i] = s;
}

// alpha[e] = sigmoid(ssrc[sidx[e]] + sdst[didx[e]] + ef[e,:] . atail[0:2])
__global__ void edge_alpha(const int* __restrict__ sidx, const int* __restrict__ didx,
                           const float* __restrict__ ef, const float* __restrict__ ssrc,
                           const float* __restrict__ sdst, const float* __restrict__ atail,
                           float* __restrict__ alpha, int E) {
  int e = blockIdx.x * blockDim.x + threadIdx.x;
  if (e >= E) return;
  float2 f = ((const float2*)ef)[e];
  float z = ssrc[sidx[e]] + sdst[didx[e]] + f.x * atail[0] + f.y * atail[1];
  alpha[e] = 1.f / (1.f + __expf(-z));
}

// Macc[didx[e], t] += alpha[e] * Hsrc[sidx[e], t]   (64 lanes per edge, coalesced)
__global__ void edge_scatter(const int* __restrict__ sidx, const int* __restrict__ didx,
                             const float* __restrict__ alpha, const float* __restrict__ Hsrc,
                             float* __restrict__ Macc, int E) {
  long gid = (long)blockIdx.x * blockDim.x + threadIdx.x;
  if (gid >= (long)E * 64) return;
  int e = (int)(gid >> 6);
  int t = (int)(gid & 63);
  int c = sidx[e], r = didx[e];
  atomicAdd(&Macc[(size_t)r * 64 + t], alpha[e] * Hsrc[(size_t)c * 64 + t]);
}

// Fused output head: h = relu(concat(h0,h1) @ W1 + b1); logits = h@W2 + b2;
// accumulate sum(ce*mask), sum(correct*mask), sum(mask).
__global__ void out_head(const float* __restrict__ h0, const float* __restrict__ h1,
                         const float* __restrict__ W1, const float* __restrict__ b1,
                         const float* __restrict__ W2, const float* __restrict__ b2,
                         const int* __restrict__ label, const int* __restrict__ mask,
                         float* __restrict__ acc, int Nrows) {
  __shared__ float lds[WPB * 16 * 32];
  int wave = threadIdx.x >> 5;
  int lane = threadIdx.x & 31;
  int tile = blockIdx.x * WPB + wave;
  bool active = (tile * 16 < Nrows);
  float* myl = &lds[wave * 16 * 32];

  if (active) {
    int rb = tile * 16, half = lane >> 4, l16 = lane & 15;
    const float* __restrict__ x0 = h0 + (size_t)(rb + l16) * 64;
    const float* __restrict__ x1 = h1 + (size_t)(rb + l16) * 64;
    v8f c0 = {}, c1 = {};
    // concat split is uniform in k0: first 64 K's from h0, next 64 from h1.
#pragma unroll 4
    for (int k0 = 0; k0 < 64; k0 += 4) {
      int kk = k0 + 2 * half;
      float2 av = *(const float2*)(x0 + kk);
      v2f a; a.x = av.x; a.y = av.y;
      const float* w0 = W1 + (size_t)kk * 32 + l16;
      v2f b0, b1v;
      b0.x = w0[0];  b1v.x = w0[16];
      b0.y = w0[32]; b1v.y = w0[48];   // next row: +32 floats
      c0 = wmma4(a, b0, c0);
      c1 = wmma4(a, b1v, c1);
    }
#pragma unroll 4
    for (int k0 = 0; k0 < 64; k0 += 4) {
      int kk = k0 + 2 * half;
      float2 av = *(const float2*)(x1 + kk);
      v2f a; a.x = av.x; a.y = av.y;
      const float* w0 = W1 + (size_t)(64 + kk) * 32 + l16;
      v2f b0, b1v;
      b0.x = w0[0];  b1v.x = w0[16];
      b0.y = w0[32]; b1v.y = w0[48];
      c0 = wmma4(a, b0, c0);
      c1 = wmma4(a, b1v, c1);
    }
#pragma unroll
    for (int r = 0; r < 8; ++r) {
      int mr = r + 8 * half;
      myl[mr * 32 + l16]      = fmaxf(c0[r] + b1[l16],      0.f);
      myl[mr * 32 + 16 + l16] = fmaxf(c1[r] + b1[16 + l16], 0.f);
    }
  }
  __syncthreads();

  float ce = 0.f, ac = 0.f, mk = 0.f;
  if (active && lane < 16) {
    int row = tile * 16 + lane;
    float l0 = b2[0], l1 = b2[1];
#pragma unroll
    for (int k = 0; k < 32; ++k) {
      float hv = myl[lane * 32 + k];
      l0 += hv * W2[2 * k];
      l1 += hv * W2[2 * k + 1];
    }
    float m   = fmaxf(l0, l1);
    float lse = m + logf(__expf(l0 - m) + __expf(l1 - m));
    int   lab = label[row];
    float cei = lse - (lab ? l1 : l0);
    int   pred = (l1 > l0) ? 1 : 0;
    float mf = (float)mask[row];
    ce = cei * mf;
    ac = (pred == lab) ? mf : 0.f;
    mk = mf;
  }
#pragma unroll
  for (int off = 16; off > 0; off >>= 1) {
    ce += __shfl_down(ce, off);
    ac += __shfl_down(ac, off);
    mk += __shfl_down(mk, off);
  }
  if (active && lane == 0) {
    atomicAdd(&acc[0], ce);
    atomicAdd(&acc[1], ac);
    atomicAdd(&acc[2], mk);
  }
}

// 0.5 * sum of squares over the 10 conv parameter tensors -> acc[3]. Single block.
__global__ void l2_kernel(const float* a0, const float* a1, const float* a2, const float* a3,
                          const float* w0, const float* c0, const float* w1, const float* c1,
                          const float* w2, const float* c2, float* __restrict__ acc) {
  __shared__ float red[256];
  int tid = threadIdx.x;
  float s = 0.f;
  for (int i = tid; i < 130;  i += 256) { s += a0[i]*a0[i] + a1[i]*a1[i] + a2[i]*a2[i] + a3[i]*a3[i]; }
  for (int i = tid; i < 4096; i += 256) { s += w0[i]*w0[i] + w1[i]*w1[i] + w2[i]*w2[i]; }
  for (int i = tid; i < 64;   i += 256) { s += c0[i]*c0[i] + c1[i]*c1[i] + c2[i]*c2[i]; }
  red[tid] = s;
  __syncthreads();
  for (int off = 128; off > 0; off >>= 1) {
    if (tid < off) red[tid] += red[tid + off];
    __syncthreads();
  }
  if (tid == 0) acc[3] = 0.5f * red[0];
}

__global__ void zero_acc(float* __restrict__ acc) {
  if (threadIdx.x < 8) acc[threadIdx.x] = 0.f;
}

__global__ void finalize(const float* __restrict__ acc, float* __restrict__ out) {
  if (threadIdx.x == 0) {
    float inv = 1.f / acc[2];
    out[0] = acc[0] * inv + WEIGHT_DECAY * acc[3];
    out[1] = acc[1] * inv;
  }
}

extern "C" void kernel_launch(void* const* d_in, const int* in_sizes, int n_in,
                              void* d_out, int out_size, void* d_ws, size_t ws_size,
                              hipStream_t stream) {
  (void)in_sizes; (void)n_in; (void)out_size; (void)ws_size;
  // ---- inputs (setup_inputs order) ----
  const float* col_feats  = (const float*)d_in[0];
  const float* row_feats  = (const float*)d_in[1];
  // d_in[2] obj_hidden: dead (obj branch unused by loss/acc)
  const int*   vc_col_idx = (const int*)d_in[3];
  const int*   vc_row_idx = (const int*)d_in[4];
  const float* vc_ef      = (const float*)d_in[5];
  const int*   cv_row_idx = (const int*)d_in[6];
  const int*   cv_col_idx = (const int*)d_in[7];
  const float* cv_ef      = (const float*)d_in[8];
  // d_in[9] vo_ef, d_in[10] co_ef: dead
  const int*   label      = (const int*)d_in[11];
  const int*   mask       = (const int*)d_in[12];
  const float* W_col_emb  = (const float*)d_in[13];
  const float* b_col_emb  = (const float*)d_in[14];
  const float* W_row_emb  = (const float*)d_in[15];
  const float* b_row_emb  = (const float*)d_in[16];
  const float* a_vc       = (const float*)d_in[17];
  const float* a_cv       = (const float*)d_in[18];
  const float* a_vo       = (const float*)d_in[19];  // L2 only
  const float* a_co       = (const float*)d_in[20];  // L2 only
  const float* W_conv_row = (const float*)d_in[21];
  const float* b_conv_row = (const float*)d_in[22];
  const float* W_conv_col = (const float*)d_in[23];
  const float* b_conv_col = (const float*)d_in[24];
  const float* W_conv_obj = (const float*)d_in[25];  // L2 only
  const float* b_conv_obj = (const float*)d_in[26];  // L2 only
  const float* W_out1     = (const float*)d_in[27];
  const float* b_out1     = (const float*)d_in[28];
  const float* W_out2     = (const float*)d_in[29];
  const float* b_out2     = (const float*)d_in[30];
  float* out = (float*)d_out;

  // ---- workspace layout (floats) ----
  float* ws = (float*)d_ws;
  size_t o = 0;
  float* col_h0  = ws + o; o += (size_t)N_COL * HH;
  float* row_h   = ws + o; o += (size_t)N_ROW * HH;
  float* msg_row = ws + o; o += (size_t)N_ROW * HH;   // row_h + sum(alpha*col_h0)
  float* msg_col = ws + o; o += (size_t)N_COL * HH;   // col_h0 + sum(alpha*row_h1)
  float* row_h1  = ws + o; o += (size_t)N_ROW * HH;
  float* col_h1  = ws + o; o += (size_t)N_COL * HH;
  float* s_src_vc = ws + o; o += N_COL;
  float* s_dst_vc = ws + o; o += N_ROW;
  float* s_src_cv = ws + o; o += N_ROW;
  float* s_dst_cv = ws + o; o += N_COL;
  float* alpha    = ws + o; o += NE;                  // reused for both directions
  float* acc      = ws + o; o += 8;                   // ce, correct, mask, l2

  const int TB = 256;
  const int colTiles = N_COL / 16, rowTiles = N_ROW / 16;
  const int colBlocks = (colTiles + WPB - 1) / WPB;
  const int rowBlocks = (rowTiles + WPB - 1) / WPB;

  zero_acc<<<1, 32, 0, stream>>>(acc);

  // embeddings (dual-store initializes the message accumulators)
  gemm_relu64_t<57><<<colBlocks, TB, 0, stream>>>(col_feats, W_col_emb, b_col_emb,
                                                  col_h0, msg_col, N_COL);
  gemm_relu64_t<26><<<rowBlocks, TB, 0, stream>>>(row_feats, W_row_emb, b_row_emb,
                                                  row_h, msg_row, N_ROW);

  // ---- vc: cols -> rows ----
  rowdot64<<<(N_COL + TB - 1) / TB, TB, 0, stream>>>(col_h0, a_vc,      s_src_vc, N_COL);
  rowdot64<<<(N_ROW + TB - 1) / TB, TB, 0, stream>>>(row_h,  a_vc + HH, s_dst_vc, N_ROW);
  edge_alpha<<<(NE + TB - 1) / TB, TB, 0, stream>>>(vc_col_idx, vc_row_idx, vc_ef,
                                                    s_src_vc, s_dst_vc, a_vc + 2 * HH,
                                                    alpha, NE);
  edge_scatter<<<(int)(((long)NE * 64 + TB - 1) / TB), TB, 0, stream>>>(
      vc_col_idx, vc_row_idx, alpha, col_h0, msg_row, NE);
  gemm_relu64_t<64><<<rowBlocks, TB, 0, stream>>>(msg_row, W_conv_row, b_conv_row,
                                                  row_h1, nullptr, N_ROW);

  // ---- cv: rows -> cols ----
  rowdot64<<<(N_ROW + TB - 1) / TB, TB, 0, stream>>>(row_h1, a_cv,      s_src_cv, N_ROW);
  rowdot64<<<(N_COL + TB - 1) / TB, TB, 0, stream>>>(col_h0, a_cv + HH, s_dst_cv, N_COL);
  edge_alpha<<<(NE + TB - 1) / TB, TB, 0, stream>>>(cv_row_idx, cv_col_idx, cv_ef,
                                                    s_src_cv, s_dst_cv, a_cv + 2 * HH,
                                                    alpha, NE);
  edge_scatter<<<(int)(((long)NE * 64 + TB - 1) / TB), TB, 0, stream>>>(
      cv_row_idx, cv_col_idx, alpha, row_h1, msg_col, NE);
  gemm_relu64_t<64><<<colBlocks, TB, 0, stream>>>(msg_col, W_conv_col, b_conv_col,
                                                  col_h1, nullptr, N_COL);

  // ---- output head + loss terms ----
  out_head<<<colBlocks, TB, 0, stream>>>(col_h0, col_h1, W_out1, b_out1, W_out2, b_out2,
                                         label, mask, acc, N_COL);
  l2_kernel<<<1, 256, 0, stream>>>(a_vc, a_cv, a_vo, a_co,
                                   W_conv_row, b_conv_row,
                                   W_conv_col, b_conv_col,
                                   W_conv_obj, b_conv_obj, acc);
  finalize<<<1, 32, 0, stream>>>(acc, out);
}